// Gimbaless_Rotation_Layer_Torch_76020921139523
// MI455X (gfx1250) — compile-verified
//
#include <hip/hip_runtime.h>

typedef __attribute__((ext_vector_type(2))) float v2f;
typedef __attribute__((ext_vector_type(8))) float v8f;

#define IMG_H 1024
#define IMG_W 1024
#define NT 32
#define CROP 4
#define TILE_W 256
#define ROWS_PER 127
#define XTILES 4
#define YCHUNKS 8
#define BLOCKS_PER_T (XTILES * YCHUNKS)
#define PART_STRIDE 12

// -------- main streaming reduction kernel --------
__global__ __launch_bounds__(256) void gimbaless_main(
    const float* __restrict__ inp, const float* __restrict__ ref,
    const float* __restrict__ lp, float* __restrict__ part)
{
  const int t     = blockIdx.y;
  const int tile  = blockIdx.x & (XTILES - 1);
  const int chunk = blockIdx.x >> 2;
  const int tid   = threadIdx.x;
  const int lane  = tid & 31;
  const int wave  = tid >> 5;

  const int  cx0   = CROP + tile * TILE_W;     // first output column of tile
  const int  x     = cx0 + tid;                // this thread's output column
  const bool valid = (x <= IMG_W - 1 - CROP);  // x in [4, 1019]
  const int  y0    = CROP + chunk * ROWS_PER;  // first output row of chunk

  __shared__ float difR[2][TILE_W + 8];
  __shared__ float avgR[2][TILE_W + 8];
  __shared__ float wsum[8][9];

  // 9-tap lowpass coefficients (uniform -> scalar loads)
  const float l0 = lp[0], l1 = lp[1], l2 = lp[2], l3 = lp[3], l4 = lp[4],
              l5 = lp[5], l6 = lp[6], l7 = lp[7], l8 = lp[8];

  const size_t base = (size_t)t * (size_t)(IMG_H * IMG_W);

  // vertical ring buffers (index 0 = oldest). After pushing row r:
  //  difc[i] = dif(r-8+i, x), avgc[i] = avg(r-8+i, x),
  //  hl[i]   = horiz lowpass of dif at row r-4+i, hg[i] = horiz grad of avg at row r-4+i
  float difc[9], avgc[9], hl[5], hg[5];
#pragma unroll
  for (int i = 0; i < 9; ++i) { difc[i] = 0.f; avgc[i] = 0.f; }
#pragma unroll
  for (int i = 0; i < 5; ++i) { hl[i] = 0.f; hg[i] = 0.f; }

  float a0 = 0.f, a1 = 0.f, a2 = 0.f, a3 = 0.f, a4 = 0.f,
        a5 = 0.f, a6 = 0.f, a7 = 0.f, a8 = 0.f;

  const float Xf = (float)x - 511.5f;  // x_vec: (x-4) - 507.5

  for (int r = y0 - 4; r <= y0 + ROWS_PER + 3; ++r) {
    const int buf = r & 1;
    {
      const size_t rowb = base + (size_t)r * IMG_W;
      const int gx = cx0 - 4 + tid;                 // [cx0-4, cx0+251]
      const float va = (gx < IMG_W) ? inp[rowb + gx] : 0.f;
      const float vb = (gx < IMG_W) ? ref[rowb + gx] : 0.f;
      difR[buf][tid] = va - vb;
      avgR[buf][tid] = 0.5f * (va + vb);
      if (tid < 8) {
        const int j   = TILE_W + tid;               // [256, 263]
        const int gx2 = cx0 - 4 + j;
        const float wa = (gx2 < IMG_W) ? inp[rowb + gx2] : 0.f;
        const float wb = (gx2 < IMG_W) ? ref[rowb + gx2] : 0.f;
        difR[buf][j] = wa - wb;
        avgR[buf][j] = 0.5f * (wa + wb);
      }
    }
    __syncthreads();

    const float* dR = difR[buf];
    const float* aR = avgR[buf];
    const int c = tid + 4;
    const float nhl = l0 * dR[c - 4] + l1 * dR[c - 3] + l2 * dR[c - 2] +
                      l3 * dR[c - 1] + l4 * dR[c]     + l5 * dR[c + 1] +
                      l6 * dR[c + 2] + l7 * dR[c + 3] + l8 * dR[c + 4];
    const float nhg = 0.5f * (aR[c - 1] - aR[c + 1]);
    const float ndv = dR[c];
    const float nav = aR[c];

    // shift rings, push newest (row r)
#pragma unroll
    for (int i = 0; i < 8; ++i) { difc[i] = difc[i + 1]; avgc[i] = avgc[i + 1]; }
    difc[8] = ndv; avgc[8] = nav;
#pragma unroll
    for (int i = 0; i < 4; ++i) { hl[i] = hl[i + 1]; hg[i] = hg[i + 1]; }
    hl[4] = nhl; hg[4] = nhg;

    if (r >= y0 + 4) {               // uniform branch: rings full, output row y = r-4
      const int y = r - 4;
      const float ABty = l0 * difc[0] + l1 * difc[1] + l2 * difc[2] +
                         l3 * difc[3] + l4 * difc[4] + l5 * difc[5] +
                         l6 * difc[6] + l7 * difc[7] + l8 * difc[8];
      const float py   = 0.5f * (avgc[3] - avgc[5]);  // rows y-1, y+1
      const float px   = hg[0];                        // row y
      const float ABtx = hl[0];                        // row y
      const float Yf   = (float)y - 511.5f;
      const float mom  = px * Yf - py * Xf;
      if (valid) {
        a0 += px * px;   a1 += px * py;   a2 += px * mom;
        a3 += py * py;   a4 += py * mom;  a5 += mom * mom;
        a6 += ABtx * px; a7 += ABty * py;
        a8 += ABtx * px * Yf - ABty * py * Xf;
      }
    }
  }

  // ---- wave-level reduction via V_WMMA_F32_16X16X4_F32 (A = ones) ----
  // With A all-ones, D[m][n] = sum_k B[k][n]; column n of B always consists of
  // {B.v0 lane n, B.v0 lane n+16, B.v1 lane n, B.v1 lane n+16}, so with v1=0:
  // every lane l of D[0] holds S_{l&15} = x_{l&15} + x_{(l&15)+16}.
  float accs[9] = {a0, a1, a2, a3, a4, a5, a6, a7, a8};
  const v2f ones = {1.0f, 1.0f};
#pragma unroll
  for (int i = 0; i < 9; ++i) {
    v2f b = {accs[i], 0.0f};
    v8f cz = {};
    v8f d = __builtin_amdgcn_wmma_f32_16x16x4_f32(
        /*neg_a=*/false, ones, /*neg_b=*/false, b,
        /*c_mod=*/(short)0, cz, /*reuse_a=*/false, /*reuse_b=*/false);
    float s = d[0];
    s += __shfl_xor(s, 1, 32);
    s += __shfl_xor(s, 2, 32);
    s += __shfl_xor(s, 4, 32);
    s += __shfl_xor(s, 8, 32);
    if (lane == 0) wsum[wave][i] = s;
  }
  __syncthreads();

  if (tid < 9) {
    float s = 0.f;
#pragma unroll
    for (int w = 0; w < 8; ++w) s += wsum[w][tid];
    part[(size_t)(t * BLOCKS_PER_T + blockIdx.x) * PART_STRIDE + tid] = s;
  }
}

// -------- final per-frame 3x3 solve (deterministic sum over block partials) --------
__global__ void gimbaless_solve(const float* __restrict__ part, float* __restrict__ out)
{
  const int t = threadIdx.x;
  if (t >= NT) return;

  double s[9];
#pragma unroll
  for (int i = 0; i < 9; ++i) s[i] = 0.0;
  for (int b = 0; b < BLOCKS_PER_T; ++b) {
    const float* p = part + (size_t)(t * BLOCKS_PER_T + b) * PART_STRIDE;
#pragma unroll
    for (int i = 0; i < 9; ++i) s[i] += (double)p[i];
  }

  const double a00 = s[0], a01 = s[1], a02 = s[2];
  const double a11 = s[3], a12 = s[4], a22 = s[5];
  const double r0 = s[6], r1 = s[7], r2 = s[8];

  const double c00 = a11 * a22 - a12 * a12;
  const double c01 = a02 * a12 - a01 * a22;
  const double c02 = a01 * a12 - a02 * a11;
  const double det = a00 * c00 + a01 * c01 + a02 * c02;
  const double id  = 1.0 / det;

  const double i00 = c00 * id, i01 = c01 * id, i02 = c02 * id;
  const double i11 = (a00 * a22 - a02 * a02) * id;
  const double i12 = (a02 * a01 - a00 * a12) * id;
  const double i22 = (a00 * a11 - a01 * a01) * id;

  out[t]          = (float)(i00 * r0 + i01 * r1 + i02 * r2);
  out[NT + t]     = (float)(i01 * r0 + i11 * r1 + i12 * r2);
  out[2 * NT + t] = (float)(i02 * r0 + i12 * r1 + i22 * r2);
}

extern "C" void kernel_launch(void* const* d_in, const int* in_sizes, int n_in,
                              void* d_out, int out_size, void* d_ws, size_t ws_size,
                              hipStream_t stream) {
  (void)in_sizes; (void)n_in; (void)out_size; (void)ws_size;
  const float* inp = (const float*)d_in[0];   // (1,32,1,1024,1024) f32
  const float* ref = (const float*)d_in[1];   // (1,32,1,1024,1024) f32
  const float* lp  = (const float*)d_in[2];   // (9,) f32
  float* part = (float*)d_ws;                 // 32*32*12 floats = 48 KB
  float* out  = (float*)d_out;                // 96 f32

  dim3 grid(BLOCKS_PER_T, NT);
  gimbaless_main<<<grid, 256, 0, stream>>>(inp, ref, lp, part);
  gimbaless_solve<<<1, 32, 0, stream>>>(part, out);
}